// SequentialCrossSpectralDensityLayer_pyriemann_76081050681796
// MI455X (gfx1250) — compile-verified
//
#include <hip/hip_runtime.h>
#include <math.h>

typedef __attribute__((ext_vector_type(2))) float v2f;
typedef __attribute__((ext_vector_type(8))) float v8f;

#define C_DIM 16
#define T_DIM 160
#define NITERS 10
#define EPSV 1e-8f
#define XS_STRIDE 162   // even & stride%64=34 -> conflict-free over 16 rows, float2-aligned
#define AUG_STRIDE 33
#define JS 17
#define NSWEEP 8

__device__ __forceinline__ v8f wmma_f32_16x16x4(v2f a, v2f b, v8f c) {
  // D = A(16x4 f32) * B(4x16 f32) + C(16x16 f32), wave32
  return __builtin_amdgcn_wmma_f32_16x16x4_f32(false, a, false, b, (short)0, c,
                                               false, false);
}

__device__ __forceinline__ float wave_sum32(float v, float* red, int lane) {
  red[lane] = v;
  __syncthreads();
  float s = 0.f;
#pragma unroll
  for (int i = 0; i < 32; ++i) s += red[i];   // broadcast reads
  __syncthreads();
  return s;
}

__device__ __forceinline__ float diag_partial(const v8f& a, int lane) {
  // acc layout: lane owns (m = r + 8*(lane>>4), n = lane&15) for r=0..7
  float p = 0.f;
  const int mb = (lane >> 4) * 8;
  const int n = lane & 15;
#pragma unroll
  for (int r = 0; r < 8; ++r) p += (n == (r + mb)) ? a[r] : 0.f;
  return p;
}

__global__ __launch_bounds__(32) void tyler_tangent_kernel(
    const float* __restrict__ x, float* __restrict__ out) {
  const int b    = blockIdx.x;
  const int lane = threadIdx.x;
  const int row  = lane & 15;
  const int hi   = lane >> 4;   // 0: lanes 0-15, 1: lanes 16-31
  const int koff = hi * 2;      // K-offset of this lane half in a 16x4 fragment

  __shared__ float Xs[C_DIM * XS_STRIDE];
  __shared__ float aug[C_DIM * AUG_STRIDE];  // [Ck+epsI | I] -> Cinv in cols 16..31
  __shared__ float As[C_DIM * JS];           // Jacobi working matrix
  __shared__ float Vs[C_DIM * JS];           // eigenvectors
  __shared__ float wbuf[T_DIM];
  __shared__ float lw[C_DIM];
  __shared__ float red[32];

  // ---------- stage X (16x160) into LDS ----------
  {
    const float4* xin =
        reinterpret_cast<const float4*>(x + (size_t)b * (C_DIM * T_DIM));
#pragma unroll
    for (int i = 0; i < 20; ++i) {
      int idx4 = lane + 32 * i;          // 640 float4 total
      float4 v = xin[idx4];
      int c = idx4 / 40;                 // 40 float4 per row (T=160)
      int t0 = (idx4 % 40) * 4;
      float* p = &Xs[c * XS_STRIDE + t0];
      p[0] = v.x; p[1] = v.y; p[2] = v.z; p[3] = v.w;
    }
  }
  __syncthreads();

  // per-lane fragment base: lane reads X[row][t + koff + {0,1}]
  const float* fragbase = &Xs[row * XS_STRIDE + koff];

  // ---------- S = X X^T via 40 chunked f32 WMMAs ----------
  // Gram symmetry: A-frag and B-frag are identical lane layouts.
  // Unroll by 4 so ds_load_b64s batch and overlap the WMMA pipeline.
  v8f acc = {0.f, 0.f, 0.f, 0.f, 0.f, 0.f, 0.f, 0.f};
#pragma unroll 1
  for (int cb = 0; cb < 10; ++cb) {
    v2f f[4];
#pragma unroll
    for (int u = 0; u < 4; ++u) {
      const float* p = fragbase + (cb * 4 + u) * 4;
      f[u].x = p[0];
      f[u].y = p[1];
    }
#pragma unroll
    for (int u = 0; u < 4; ++u) acc = wmma_f32_16x16x4(f[u], f[u], acc);
  }
  // Ck = (S/T) * C / (trace(S/T)+eps)
  float tr = wave_sum32(diag_partial(acc, lane) * (1.f / T_DIM), red, lane);
  float scl = (1.f / T_DIM) * (16.f / (tr + EPSV));
  v8f ck;
#pragma unroll
  for (int r = 0; r < 8; ++r) ck[r] = acc[r] * scl;

  // ---------- Tyler fixed-point iterations ----------
#pragma unroll 1
  for (int it = 0; it < NITERS; ++it) {
    // write augmented [Ck + eps*I | I]
#pragma unroll
    for (int r = 0; r < 8; ++r) {
      int m = r + 8 * hi;
      aug[m * AUG_STRIDE + row]      = ck[r] + ((m == row) ? EPSV : 0.f);
      aug[m * AUG_STRIDE + 16 + row] = (m == row) ? 1.f : 0.f;
    }
    __syncthreads();

    // Gauss-Jordan inverse: 16 pivots, 32 columns handled by 32 lanes
#pragma unroll 1
    for (int k = 0; k < 16; ++k) {
      float pinv = 1.f / aug[k * AUG_STRIDE + k];
      __syncthreads();
      aug[k * AUG_STRIDE + lane] *= pinv;
      __syncthreads();
#pragma unroll 1
      for (int r = 0; r < 16; ++r) {
        if (r == k) continue;
        float fct = aug[r * AUG_STRIDE + k];
        aug[r * AUG_STRIDE + lane] -= fct * aug[k * AUG_STRIDE + lane];
      }
      __syncthreads();
    }

    // q_t = x_t^T Cinv x_t, w_t = C/max(q,eps); each lane does 5 columns
#pragma unroll 1
    for (int j5 = 0; j5 < 5; ++j5) {
      int t = lane + 32 * j5;
      float xc[16];
#pragma unroll
      for (int i = 0; i < 16; ++i) xc[i] = Xs[i * XS_STRIDE + t];
      float qv = 0.f;
#pragma unroll
      for (int i = 0; i < 16; ++i) {
        const float* ci = &aug[i * AUG_STRIDE + 16];  // broadcast row of Cinv
        qv += ci[i] * xc[i] * xc[i];
        float ra = 0.f;
#pragma unroll
        for (int j = i + 1; j < 16; ++j) ra += ci[j] * xc[j];
        qv += 2.f * xc[i] * ra;
      }
      wbuf[t] = 16.f / fmaxf(qv, EPSV);
    }
    __syncthreads();

    // Cn = (w .* X) X^T via WMMA: weights folded into A fragment only
    v8f a2 = {0.f, 0.f, 0.f, 0.f, 0.f, 0.f, 0.f, 0.f};
#pragma unroll 1
    for (int cb = 0; cb < 10; ++cb) {
      v2f bb[4], aa[4];
#pragma unroll
      for (int u = 0; u < 4; ++u) {
        int t = (cb * 4 + u) * 4 + koff;
        const float* p = fragbase + (cb * 4 + u) * 4;
        bb[u].x = p[0];
        bb[u].y = p[1];
        aa[u].x = bb[u].x * wbuf[t];       // even-aligned pair -> mergeable load
        aa[u].y = bb[u].y * wbuf[t + 1];
      }
#pragma unroll
      for (int u = 0; u < 4; ++u) a2 = wmma_f32_16x16x4(aa[u], bb[u], a2);
    }
    float tr2 = wave_sum32(diag_partial(a2, lane) * (1.f / T_DIM), red, lane);
    float s2 = (1.f / T_DIM) * (16.f / (tr2 + EPSV));
#pragma unroll
    for (int r = 0; r < 8; ++r) ck[r] = a2[r] * s2;
    __syncthreads();
  }

  // ---------- symmetrize 0.5*(C + C^T) ----------
#pragma unroll
  for (int r = 0; r < 8; ++r) { int m = r + 8 * hi; As[m * JS + row] = ck[r]; }
  __syncthreads();
  float tv[8];
#pragma unroll
  for (int r = 0; r < 8; ++r) { int m = r + 8 * hi; tv[r] = As[row * JS + m]; }
  __syncthreads();
#pragma unroll
  for (int r = 0; r < 8; ++r) {
    int m = r + 8 * hi;
    As[m * JS + row] = 0.5f * (ck[r] + tv[r]);
  }
  if (lane < 16) {
#pragma unroll
    for (int j = 0; j < 16; ++j) Vs[lane * JS + j] = (lane == j) ? 1.f : 0.f;
  }
  __syncthreads();

  // ---------- cyclic Jacobi eigh (16x16 symmetric) ----------
#pragma unroll 1
  for (int sweep = 0; sweep < NSWEEP; ++sweep) {
#pragma unroll 1
    for (int p = 0; p < 15; ++p) {
#pragma unroll 1
      for (int q = p + 1; q < 16; ++q) {
        float apq = As[p * JS + q];
        float app = As[p * JS + p];
        float aqq = As[q * JS + q];
        float cc = 1.f, ss = 0.f;
        if (fabsf(apq) > 1e-30f) {            // uniform across lanes
          float th = (aqq - app) / (2.f * apq);
          float tt = 1.f / (fabsf(th) + sqrtf(th * th + 1.f));
          if (th < 0.f) tt = -tt;
          cc = 1.f / sqrtf(1.f + tt * tt);
          ss = tt * cc;
        }
        __syncthreads();
        int k2 = lane & 15;
        if (lane < 16) {                      // rotate A columns p,q
          float akp = As[k2 * JS + p], akq = As[k2 * JS + q];
          As[k2 * JS + p] = cc * akp - ss * akq;
          As[k2 * JS + q] = ss * akp + cc * akq;
        } else {                              // rotate V columns p,q in parallel
          float vkp = Vs[k2 * JS + p], vkq = Vs[k2 * JS + q];
          Vs[k2 * JS + p] = cc * vkp - ss * vkq;
          Vs[k2 * JS + q] = ss * vkp + cc * vkq;
        }
        __syncthreads();
        if (lane < 16) {                      // rotate A rows p,q
          float apk = As[p * JS + k2], aqk = As[q * JS + k2];
          As[p * JS + k2] = cc * apk - ss * aqk;
          As[q * JS + k2] = ss * apk + cc * aqk;
        }
        __syncthreads();
      }
    }
  }
  if (lane < 16) lw[lane] = logf(fmaxf(As[lane * JS + lane], EPSV));
  __syncthreads();

  // ---------- logC = V diag(logw) V^T via 4 WMMAs (K=16) ----------
  v8f lg = {0.f, 0.f, 0.f, 0.f, 0.f, 0.f, 0.f, 0.f};
#pragma unroll
  for (int kc = 0; kc < 4; ++kc) {
    int k0 = kc * 4 + koff;
    v2f bb; bb.x = Vs[row * JS + k0]; bb.y = Vs[row * JS + k0 + 1];
    float l0 = lw[k0], l1 = lw[k0 + 1];
    v2f aa; aa.x = bb.x * l0; aa.y = bb.y * l1;
    lg = wmma_f32_16x16x4(aa, bb, lg);
  }

  // ---------- scatter upper triangle with sqrt(2) off-diagonal ----------
  const float SQ2 = 1.4142135623730951f;
  float* op = out + (size_t)b * 136;
#pragma unroll
  for (int r = 0; r < 8; ++r) {
    int m = r + 8 * hi;        // this lane owns logC[m][row]
    if (m <= row) {
      int idx = m * 16 - (m * (m - 1)) / 2 + (row - m);  // row-major triu index
      op[idx] = lg[r] * ((m == row) ? 1.f : SQ2);
    }
  }
}

extern "C" void kernel_launch(void* const* d_in, const int* in_sizes, int n_in,
                              void* d_out, int out_size, void* d_ws,
                              size_t ws_size, hipStream_t stream) {
  const float* x = (const float*)d_in[0];
  float* out = (float*)d_out;
  int B = in_sizes[0] / (C_DIM * T_DIM);   // 8192 for the reference shapes
  tyler_tangent_kernel<<<B, 32, 0, stream>>>(x, out);
}